// RepSharingKernelNet_46651934769564
// MI455X (gfx1250) — compile-verified
//
#include <hip/hip_runtime.h>
#include <math.h>
#include <stdint.h>

// ---------------------------------------------------------------------------
// Fused guided box-filter combine for MI455X (gfx1250).
// Bandwidth-bound (~336 MB min traffic -> ~14us @ 23.3 TB/s), so: single fused
// pass, TDM (tensor_load_to_lds) stages haloed tiles into LDS, split
// s_wait_tensorcnt overlaps guide-exp with irradiance DMA. VALU sliding sums
// (not WMMA) since f32 stencil compute is far below memory time.
// Toolchain note: 6-arg tensor_load_to_lds (amdgpu-toolchain / clang-23 form).
// ---------------------------------------------------------------------------

#define HH   1024
#define WW   1024
#define NB   4
#define TX   32            // tile width  (outputs)
#define TY   16            // tile height (outputs)
#define PMAX 6             // max box radius (k=13)
#define RX   (TX + 2*PMAX) // 44 haloed cols
#define RY   (TY + 2*PMAX) // 28 haloed rows
#define NK   6
#define HW   (HH*WW)

typedef unsigned int v4u __attribute__((ext_vector_type(4)));
typedef int          v8i __attribute__((ext_vector_type(8)));
typedef int          v4i __attribute__((ext_vector_type(4)));

// Issue one TDM 2D tile load: RX x RY f32 tile, row stride WW elements,
// packed contiguously into LDS at lds_byte_off (pitch = RX floats).
__device__ __forceinline__ void tdm_load_tile(uint32_t lds_byte_off,
                                              const float* gsrc) {
  uint64_t ga = (uint64_t)(uintptr_t)gsrc;

  v4u g0;
  g0.x = 1u;                                           // count=1, user desc
  g0.y = lds_byte_off;                                 // lds_addr (bytes)
  g0.z = (uint32_t)ga;                                 // global_addr[31:0]
  g0.w = (uint32_t)((ga >> 32) & 0x01FFFFFFu)          // global_addr[56:32]
       | (2u << 30);                                   // type=2 ("image")

  const uint32_t TD0 = 1u << 30;                       // huge tensor dims:
  const uint32_t TD1 = 1u << 30;                       // interior tiles never OOB
  v8i g1;
  g1[0] = (int)(2u << 16);                             // data_size=4B; mask=0
  g1[1] = (int)((TD0 & 0xFFFFu) << 16);                // tensor_dim0 lo16
  g1[2] = (int)((TD0 >> 16) | ((TD1 & 0xFFFFu) << 16));// td0 hi16 | td1 lo16
  g1[3] = (int)((TD1 >> 16) | ((uint32_t)RX << 16));   // td1 hi16 | tile_dim0=44
  g1[4] = (int)(uint32_t)RY;                           // tile_dim1=28, tile_dim2=0
  g1[5] = (int)(uint32_t)WW;                           // tensor_dim0_stride = 1024
  g1[6] = 0;                                           // stride0 hi | stride1 lo
  g1[7] = 0;

  v4i z4; z4[0]=0; z4[1]=0; z4[2]=0; z4[3]=0;          // groups 2/3 unused (2D)
  v8i z8; z8[0]=0; z8[1]=0; z8[2]=0; z8[3]=0;
          z8[4]=0; z8[5]=0; z8[6]=0; z8[7]=0;
  __builtin_amdgcn_tensor_load_to_lds(g0, g1, z4, z4, z8, 0);
}

__global__ __launch_bounds__(256)
void fused_guided_box(const float* __restrict__ x_in,   // (4,6,H,W): irr 0:3, alb 3:6
                      const float* __restrict__ x_fo,   // (4,12,H,W): guide 0:6, alpha 6:12
                      float* __restrict__ out) {        // (4,3,H,W)
  __shared__ float sG[NK][RY*RX];    // guide logits -> exp(g)
  __shared__ float sIrr[3][RY*RX];   // irradiance haloed tiles
  __shared__ float sV[TY*RX];        // vertical-sum scratch (one size at a time)

  const int t  = (int)threadIdx.x;
  const int x0 = (int)blockIdx.x * TX;
  const int y0 = (int)blockIdx.y * TY;
  const int b  = (int)blockIdx.z;

  const float* foP  = x_fo + (size_t)b * 12 * HW;
  const float* inP  = x_in + (size_t)b * 6  * HW;

  const bool interior = (x0 >= PMAX) && (y0 >= PMAX) &&
                        (x0 + TX + PMAX <= WW) && (y0 + TY + PMAX <= HH);

  if (interior) {
    if (t == 0) {
      const int base = (y0 - PMAX) * WW + (x0 - PMAX);
      #pragma unroll
      for (int i = 0; i < NK; ++i)
        tdm_load_tile((uint32_t)(uintptr_t)&sG[i][0], foP + (size_t)i * HW + base);
      #pragma unroll
      for (int c = 0; c < 3; ++c)
        tdm_load_tile((uint32_t)(uintptr_t)&sIrr[c][0], inP + (size_t)c * HW + base);
    }
  } else {
    // Border tiles: bounds-checked loads; OOB logit -> -1e30 (exp -> 0 == zero pad).
    for (int idx = t; idx < NK * RY * RX; idx += 256) {
      int ch = idx / (RY * RX);
      int r  = (idx / RX) % RY;
      int cx = idx % RX;
      int gy = y0 - PMAX + r, gx = x0 - PMAX + cx;
      bool ok = (gy >= 0) & (gy < HH) & (gx >= 0) & (gx < WW);
      sG[ch][r * RX + cx] = ok ? foP[(size_t)ch * HW + (size_t)gy * WW + gx] : -1e30f;
    }
    for (int idx = t; idx < 3 * RY * RX; idx += 256) {
      int ch = idx / (RY * RX);
      int r  = (idx / RX) % RY;
      int cx = idx % RX;
      int gy = y0 - PMAX + r, gx = x0 - PMAX + cx;
      bool ok = (gy >= 0) & (gy < HH) & (gx >= 0) & (gx < WW);
      sIrr[ch][r * RX + cx] = ok ? inP[(size_t)ch * HW + (size_t)gy * WW + gx] : 0.0f;
    }
  }

  // First 6 TDM loads (guide planes) done; irradiance DMA still in flight.
  __builtin_amdgcn_s_wait_tensorcnt(3);
  __syncthreads();

  // g = exp(logit) in place (overlaps with irradiance TDM).
  {
    float* gflat = &sG[0][0];
    for (int idx = t; idx < NK * RY * RX; idx += 256)
      gflat[idx] = __expf(gflat[idx]);
  }

  // Per-pixel alpha softmax (2 output pixels per thread), kept in registers.
  float aw[2][NK];
  int   py[2], px[2];
  #pragma unroll
  for (int q = 0; q < 2; ++q) {
    int l  = t + q * 256;
    int ty = l / TX, tx = l % TX;
    py[q] = y0 + ty; px[q] = x0 + tx;
    size_t o = (size_t)py[q] * WW + px[q];
    float m[NK]; float mx = -1e30f;
    #pragma unroll
    for (int i = 0; i < NK; ++i) { m[i] = foP[(size_t)(NK + i) * HW + o]; mx = fmaxf(mx, m[i]); }
    float s = 0.0f;
    #pragma unroll
    for (int i = 0; i < NK; ++i) { m[i] = __expf(m[i] - mx); s += m[i]; }
    float inv = 1.0f / s;
    #pragma unroll
    for (int i = 0; i < NK; ++i) aw[q][i] = m[i] * inv;
  }

  // Denominators: per size, vertical k-sum of g into sV, horizontal k-sum per
  // pixel, then fold alpha: aw[q][i] := alpha_i / den_i.
  #pragma unroll
  for (int i = 0; i < NK; ++i) {
    const int p = i + 1;                 // radius: k = 2p+1 in {3..13}
    __syncthreads();                      // sG ready / sV reusable
    for (int idx = t; idx < TY * RX; idx += 256) {
      int r = idx / RX, cx = idx % RX;
      const float* col = &sG[i][(r + PMAX - p) * RX + cx];
      float s = 0.0f;
      #pragma unroll
      for (int dy = 0; dy < 2 * p + 1; ++dy) s += col[dy * RX];
      sV[idx] = s;
    }
    __syncthreads();
    #pragma unroll
    for (int q = 0; q < 2; ++q) {
      int l  = t + q * 256;
      int ty = l / TX, tx = l % TX;
      const float* row = &sV[ty * RX + tx + PMAX - p];
      float den = 0.0f;
      #pragma unroll
      for (int dx = 0; dx < 2 * p + 1; ++dx) den += row[dx];
      aw[q][i] = aw[q][i] / den;
    }
  }

  // Irradiance DMA must be complete before numerator phase.
  __builtin_amdgcn_s_wait_tensorcnt(0);
  __syncthreads();

  // Numerators per color channel; accumulate sum_i w_i * box(g_i * irr_c).
  #pragma unroll
  for (int c = 0; c < 3; ++c) {
    float acc[2] = {0.0f, 0.0f};
    #pragma unroll
    for (int i = 0; i < NK; ++i) {
      const int p = i + 1;
      __syncthreads();                    // sV reuse fence
      for (int idx = t; idx < TY * RX; idx += 256) {
        int r = idx / RX, cx = idx % RX;
        int o = (r + PMAX - p) * RX + cx;
        float s = 0.0f;
        #pragma unroll
        for (int dy = 0; dy < 2 * p + 1; ++dy)
          s += sG[i][o + dy * RX] * sIrr[c][o + dy * RX];
        sV[idx] = s;
      }
      __syncthreads();
      #pragma unroll
      for (int q = 0; q < 2; ++q) {
        int l  = t + q * 256;
        int ty = l / TX, tx = l % TX;
        const float* row = &sV[ty * RX + tx + PMAX - p];
        float num = 0.0f;
        #pragma unroll
        for (int dx = 0; dx < 2 * p + 1; ++dx) num += row[dx];
        acc[q] += aw[q][i] * num;
      }
    }
    #pragma unroll
    for (int q = 0; q < 2; ++q) {
      size_t o   = (size_t)py[q] * WW + px[q];
      float  alb = inP[(size_t)(3 + c) * HW + o];
      out[(size_t)(b * 3 + c) * HW + o] = acc[q] * alb;
    }
  }
}

extern "C" void kernel_launch(void* const* d_in, const int* in_sizes, int n_in,
                              void* d_out, int out_size, void* d_ws, size_t ws_size,
                              hipStream_t stream) {
  (void)in_sizes; (void)n_in; (void)out_size; (void)d_ws; (void)ws_size;
  const float* x_in = (const float*)d_in[0];   // (4,6,1024,1024)
  const float* x_fo = (const float*)d_in[1];   // (4,12,1024,1024)
  float*       out  = (float*)d_out;           // (4,3,1024,1024)

  dim3 grid(WW / TX, HH / TY, NB);             // 32 x 64 x 4 tiles
  dim3 block(256);                             // 8 wave32 waves
  fused_guided_box<<<grid, block, 0, stream>>>(x_in, x_fo, out);
}